// ChebConv_56908316672630
// MI455X (gfx1250) — compile-verified
//
#include <hip/hip_runtime.h>

typedef __attribute__((ext_vector_type(2))) float v2f;
typedef __attribute__((ext_vector_type(8))) float v8f;

#define D 128          // feature dim (in == out == 128)
#define LDSPAD 132     // padded LDS row stride (132 % 64 banks -> conflict-free)

// ---- degree: deg[src] += (src != dst) -------------------------------------
__global__ void cheb_deg_kernel(const int* __restrict__ ei, int E,
                                float* __restrict__ deg) {
    int e = blockIdx.x * blockDim.x + threadIdx.x;
    if (e >= E) return;
    int s = ei[e];
    int d = ei[E + e];
    if (s != d) unsafeAtomicAdd(&deg[s], 1.0f);
}

// ---- dis[i] = deg>0 ? rsqrt(max(deg,1)) : 0 -------------------------------
__global__ void cheb_dis_kernel(const float* __restrict__ deg,
                                float* __restrict__ dis, int n) {
    int i = blockIdx.x * blockDim.x + threadIdx.x;
    if (i >= n) return;
    float dv = deg[i];
    dis[i] = (dv > 0.0f) ? rsqrtf(fmaxf(dv, 1.0f)) : 0.0f;
}

// ---- SpMM: agg[dst] += -(dis[src]*dis[dst]) * h[src]  (one wave per edge) --
__global__ void cheb_spmm_kernel(const int* __restrict__ ei, int E,
                                 const float* __restrict__ dis,
                                 const float* __restrict__ hin,
                                 float* __restrict__ agg) {
    int wid  = (int)((blockIdx.x * (long)blockDim.x + threadIdx.x) >> 5);
    int lane = threadIdx.x & 31;
    if (wid >= E) return;
    int s = ei[wid];
    int d = ei[E + wid];
    float w = (s != d) ? -(dis[s] * dis[d]) : 0.0f;   // uniform across wave
    if (w == 0.0f) return;
    int c = lane * 4;
    const float4 h = *(const float4*)(hin + (long)s * D + c);
    float* o = agg + (long)d * D + c;
    unsafeAtomicAdd(o + 0, w * h.x);
    unsafeAtomicAdd(o + 1, w * h.y);
    unsafeAtomicAdd(o + 2, w * h.z);
    unsafeAtomicAdd(o + 3, w * h.w);
}

// ---- fused GEMM: out = Tx0@W0 + Tx1@W1 + (2S - Tx0)@W2 + bias -------------
// 256 threads = 8 waves per block; block owns 16 node rows; wave w owns
// output columns [16w, 16w+16). FP32 WMMA 16x16x4, k-loop over D=128.
__global__ void __launch_bounds__(256)
cheb_gemm_kernel(const float* __restrict__ x,
                 const float* __restrict__ tx1,
                 const float* __restrict__ s2,
                 const float* __restrict__ weight,   // [3][128][128]
                 const float* __restrict__ bias,     // [128]
                 float* __restrict__ out, int n) {
    __shared__ float lA[3][16][LDSPAD];

    const int m0  = blockIdx.x * 16;
    const int tid = threadIdx.x;

    // cooperative LDS stage of the three 16x128 A-tiles (Tx2 fused as 2S - x)
    for (int idx = tid; idx < 3 * 16 * D; idx += 256) {
        int t = idx >> 11;         // / 2048
        int r = (idx >> 7) & 15;   // row within tile
        int c = idx & (D - 1);     // column
        int row = m0 + r;
        float v = 0.0f;
        if (row < n) {
            long off = (long)row * D + c;
            if (t == 0)      v = x[off];
            else if (t == 1) v = tx1[off];
            else             v = 2.0f * s2[off] - x[off];
        }
        lA[t][r][c] = v;
    }
    __syncthreads();

    const int wave = tid >> 5;
    const int lane = tid & 31;
    const int hf   = lane >> 4;    // lane half: selects K pair / M half
    const int lm   = lane & 15;
    const int col  = wave * 16 + lm;

    v8f acc = {};
    for (int kk = 0; kk < D; kk += 4) {
        const int ka = kk + 2 * hf;    // A/B K index held in VGPR0 this half
#pragma unroll
        for (int t = 0; t < 3; ++t) {
            // A 16x4 fragment: lane lm holds row M=lm, K = ka, ka+1
            v2f a;
            a.x = lA[t][lm][ka];
            a.y = lA[t][lm][ka + 1];
            // B 4x16 fragment: VGPR j, half hf -> row K = kk + j + 2*hf, col
            const float* W = weight + (long)t * D * D;
            v2f b;
            b.x = W[(long)ka * D + col];
            b.y = W[(long)(ka + 1) * D + col];
            acc = __builtin_amdgcn_wmma_f32_16x16x4_f32(
                false, a, false, b, (short)0, acc, false, false);
        }
    }

    const float bv = bias[col];
#pragma unroll
    for (int i = 0; i < 8; ++i) {
        int row = m0 + i + 8 * hf;     // C/D layout: VGPR i -> M = i + 8*half
        if (row < n) out[(long)row * D + col] = acc[i] + bv;
    }
}

extern "C" void kernel_launch(void* const* d_in, const int* in_sizes, int n_in,
                              void* d_out, int out_size, void* d_ws, size_t ws_size,
                              hipStream_t stream) {
    const float* x      = (const float*)d_in[0];
    const int*   ei     = (const int*)d_in[1];
    const float* weight = (const float*)d_in[2];
    const float* bias   = (const float*)d_in[3];
    float*       out    = (float*)d_out;

    const int n = in_sizes[0] / D;       // 100000
    const int E = in_sizes[1] / 2;       // 1600000

    // workspace: deg[n] | dis[n] | Tx1[n*D] | S[n*D]
    float* deg = (float*)d_ws;
    float* dis = deg + n;
    float* tx1 = dis + n;
    float* s2  = tx1 + (long)n * D;

    const size_t zbytes = ((size_t)2 * n + (size_t)2 * n * D) * sizeof(float);
    hipMemsetAsync(d_ws, 0, zbytes, stream);

    cheb_deg_kernel<<<(E + 255) / 256, 256, 0, stream>>>(ei, E, deg);
    cheb_dis_kernel<<<(n + 255) / 256, 256, 0, stream>>>(deg, dis, n);

    const int spmm_blocks = (E + 7) / 8;   // 8 waves (edges) per 256-thr block
    cheb_spmm_kernel<<<spmm_blocks, 256, 0, stream>>>(ei, E, dis, x,   tx1);
    cheb_spmm_kernel<<<spmm_blocks, 256, 0, stream>>>(ei, E, dis, tx1, s2);

    cheb_gemm_kernel<<<(n + 15) / 16, 256, 0, stream>>>(x, tx1, s2, weight,
                                                        bias, out, n);
}